// CustomLoss_28123445854450
// MI455X (gfx1250) — compile-verified
//
#include <hip/hip_runtime.h>

// ---------------------------------------------------------------------------
// SSIM(3x3 avg-pool, zero pad, /9) + L1 combined loss, (16,1,1024,1024) fp32.
// Bandwidth-bound: 134 MB reads -> ~5.8us floor @ 23.3 TB/s. Strategy:
//   * Tensor Data Mover (tensor_load_to_lds) stages the whole 130x34 halo
//     tile per buffer in ONE instruction, with hardware OOB zero-fill for the
//     right/bottom halo (tensor_dim is tile-start-relative).
//   * Tiles touching the top/left image border (negative tile start, which
//     TDM cannot express) fall back to per-lane async global->LDS DMA
//     (GLOBAL_LOAD_ASYNC_TO_LDS_B32) + explicit zero stores.
//   * separable 3x3 with a rolling 3-row register window per column,
//   * deterministic two-pass reduction (no float atomics).
// ---------------------------------------------------------------------------

#define ALPHA_W 0.85f
#define BETA_W  0.15f
#define C1_CONST (0.01f * 0.01f)
#define C2_CONST (0.03f * 0.03f)

#define IMG_H 1024
#define IMG_W 1024
#define TILE_W 128
#define TILE_H 32
#define HALO_W (TILE_W + 2)      // 130 (also the LDS row stride: TDM-contiguous)
#define HALO_H (TILE_H + 2)      // 34
#define NTHREADS 128

typedef unsigned uint4v __attribute__((ext_vector_type(4)));
typedef unsigned uint8v __attribute__((ext_vector_type(8)));

__global__ __launch_bounds__(NTHREADS)
void ssim_l1_partial_kernel(const float* __restrict__ pred,
                            const float* __restrict__ target,
                            float* __restrict__ partials)
{
    __shared__ float sp[HALO_H * HALO_W];
    __shared__ float st[HALO_H * HALO_W];
    __shared__ float red[2 * NTHREADS];

    const int tid = threadIdx.x;
    const int x0  = blockIdx.x * TILE_W;
    const int y0  = blockIdx.y * TILE_H;
    const size_t img_off = (size_t)blockIdx.z * (size_t)(IMG_W * IMG_H);
    const float* __restrict__ pimg = pred + img_off;
    const float* __restrict__ timg = target + img_off;

    const bool interior = (x0 > 0) && (y0 > 0);  // uniform per block

    if (interior) {
        // ---- TDM path: one DMA per buffer stages the whole halo tile -------
        // D# is built relative to the tile start at (y0-1, x0-1):
        //   tensor_dim0/1 = elements remaining to the image edge, so reads
        //   past the right/bottom edge return zero (== zero padding).
        if (tid == 0) {
            const size_t tile_elt = (size_t)(y0 - 1) * IMG_W + (size_t)(x0 - 1);
            const unsigned td0 = (unsigned)(IMG_W - (x0 - 1));
            const unsigned td1 = (unsigned)(IMG_H - (y0 - 1));

            auto tdm_load = [&](const float* gsrc, float* ldst) {
                const unsigned long long ga = (unsigned long long)(size_t)gsrc;
                uint4v g0;
                g0.x = 1u;                                   // count=1 (valid D#)
                g0.y = (unsigned)(size_t)ldst;               // lds_addr (byte offset)
                g0.z = (unsigned)ga;                         // global_addr[31:0]
                g0.w = ((unsigned)(ga >> 32) & 0x01FFFFFFu)  // global_addr[56:32]
                       | (2u << 30);                         // type=2 ("image")
                uint8v g1;
                g1.s0 = 0x00020000u;                         // data_size=2 (4 bytes)
                g1.s1 = (td0 & 0xFFFFu) << 16;               // tensor_dim0[15:0]
                g1.s2 = (td0 >> 16) | ((td1 & 0xFFFFu) << 16);
                g1.s3 = (td1 >> 16) | ((unsigned)HALO_W << 16);  // tile_dim0=130
                g1.s4 = (unsigned)HALO_H;                    // tile_dim1=34, tile_dim2=0
                g1.s5 = (unsigned)IMG_W;                     // tensor_dim0_stride=1024
                g1.s6 = 0u;                                  // stride hi / dim1_stride
                g1.s7 = 0u;
                asm volatile("tensor_load_to_lds %0, %1"
                             :: "s"(g0), "s"(g1) : "memory");
            };
            tdm_load(pimg + tile_elt, sp);
            tdm_load(timg + tile_elt, st);
        }
        __builtin_amdgcn_s_wait_tensorcnt(0);
        __syncthreads();
    } else {
        // ---- Fallback: per-lane async global->LDS DMA (top/left border) ----
        const int halo_elems = HALO_H * HALO_W;  // 4420 per buffer
        for (int idx = tid; idx < 2 * halo_elems; idx += NTHREADS) {
            const int which = (idx >= halo_elems) ? 1 : 0;
            const int e = which ? (idx - halo_elems) : idx;
            const int r = e / HALO_W;
            const int c = e - r * HALO_W;
            const int gy = y0 - 1 + r;
            const int gx = x0 - 1 + c;
            float* dst = (which ? st : sp) + (r * HALO_W + c);
            if ((unsigned)gy < (unsigned)IMG_H && (unsigned)gx < (unsigned)IMG_W) {
                const float* src = (which ? timg : pimg) + ((size_t)gy * IMG_W + gx);
                unsigned lds_addr = (unsigned)(size_t)dst;
                asm volatile("global_load_async_to_lds_b32 %0, %1, off"
                             :: "v"(lds_addr), "v"(src) : "memory");
            } else {
                *dst = 0.0f;  // zero padding
            }
        }
        asm volatile("s_wait_asynccnt 0x0" ::: "memory");
        __syncthreads();
    }

    // ---- Separable 3x3: rolling 3-row window of the five row-sums ---------
    const int x = tid;  // column owned by this thread (0..127)
    float ssim_acc = 0.0f;
    float l1_acc   = 0.0f;

    float Rp0, Rt0, Rpp0, Rtt0, Rpt0;
    float Rp1, Rt1, Rpp1, Rtt1, Rpt1;
    float Rp2, Rt2, Rpp2, Rtt2, Rpt2;

    auto rowsum = [&](int r, float& Rp, float& Rt,
                      float& Rpp, float& Rtt, float& Rpt) {
        const int b = r * HALO_W + x;
        const float p0 = sp[b], p1 = sp[b + 1], p2 = sp[b + 2];
        const float t0 = st[b], t1 = st[b + 1], t2 = st[b + 2];
        Rp  = p0 + p1 + p2;
        Rt  = t0 + t1 + t2;
        Rpp = fmaf(p0, p0, fmaf(p1, p1, p2 * p2));
        Rtt = fmaf(t0, t0, fmaf(t1, t1, t2 * t2));
        Rpt = fmaf(p0, t0, fmaf(p1, t1, p2 * t2));
    };

    rowsum(0, Rp0, Rt0, Rpp0, Rtt0, Rpt0);
    rowsum(1, Rp1, Rt1, Rpp1, Rtt1, Rpt1);

    const float inv9 = 1.0f / 9.0f;
    #pragma unroll 4
    for (int y = 0; y < TILE_H; ++y) {
        rowsum(y + 2, Rp2, Rt2, Rpp2, Rtt2, Rpt2);

        const float Sp  = Rp0  + Rp1  + Rp2;
        const float St  = Rt0  + Rt1  + Rt2;
        const float Spp = Rpp0 + Rpp1 + Rpp2;
        const float Stt = Rtt0 + Rtt1 + Rtt2;
        const float Spt = Rpt0 + Rpt1 + Rpt2;

        const float mu_x = Sp * inv9;
        const float mu_y = St * inv9;
        const float vx  = Spp * inv9 - mu_x * mu_x;
        const float vy  = Stt * inv9 - mu_y * mu_y;
        const float vxy = Spt * inv9 - mu_x * mu_y;

        const float nmr = (2.0f * mu_x * mu_y + C1_CONST) * (2.0f * vxy + C2_CONST);
        const float dnm = (mu_x * mu_x + mu_y * mu_y + C1_CONST) * (vx + vy + C2_CONST);
        float contrib = (1.0f - nmr / dnm) * 0.5f;
        contrib = fminf(fmaxf(contrib, 0.0f), 1.0f);
        ssim_acc += contrib;

        const float pc = sp[(y + 1) * HALO_W + x + 1];
        const float tc = st[(y + 1) * HALO_W + x + 1];
        l1_acc += fabsf(pc - tc);

        // rotate the window
        Rp0 = Rp1;  Rt0 = Rt1;  Rpp0 = Rpp1;  Rtt0 = Rtt1;  Rpt0 = Rpt1;
        Rp1 = Rp2;  Rt1 = Rt2;  Rpp1 = Rpp2;  Rtt1 = Rtt2;  Rpt1 = Rpt2;
    }

    // ---- Block reduction (deterministic) ----------------------------------
    red[tid] = ssim_acc;
    red[NTHREADS + tid] = l1_acc;
    __syncthreads();
    #pragma unroll
    for (int off = NTHREADS / 2; off > 0; off >>= 1) {
        if (tid < off) {
            red[tid] += red[tid + off];
            red[NTHREADS + tid] += red[NTHREADS + tid + off];
        }
        __syncthreads();
    }
    if (tid == 0) {
        const int bidx = (blockIdx.z * gridDim.y + blockIdx.y) * gridDim.x + blockIdx.x;
        partials[2 * bidx + 0] = red[0];
        partials[2 * bidx + 1] = red[NTHREADS];
    }
}

__global__ __launch_bounds__(256)
void ssim_l1_finalize_kernel(const float* __restrict__ partials,
                             int nblocks, float inv_n,
                             float* __restrict__ out)
{
    __shared__ float red[512];
    float s = 0.0f, l = 0.0f;
    for (int i = threadIdx.x; i < nblocks; i += 256) {
        s += partials[2 * i + 0];
        l += partials[2 * i + 1];
    }
    red[threadIdx.x] = s;
    red[256 + threadIdx.x] = l;
    __syncthreads();
    #pragma unroll
    for (int off = 128; off > 0; off >>= 1) {
        if ((int)threadIdx.x < off) {
            red[threadIdx.x] += red[threadIdx.x + off];
            red[256 + threadIdx.x] += red[256 + threadIdx.x + off];
        }
        __syncthreads();
    }
    if (threadIdx.x == 0) {
        out[0] = ALPHA_W * (red[0] * inv_n) + BETA_W * (red[256] * inv_n);
    }
}

extern "C" void kernel_launch(void* const* d_in, const int* in_sizes, int n_in,
                              void* d_out, int out_size, void* d_ws, size_t ws_size,
                              hipStream_t stream)
{
    const float* pred   = (const float*)d_in[0];
    const float* target = (const float*)d_in[1];
    float* out      = (float*)d_out;
    float* partials = (float*)d_ws;   // needs grid_total * 2 floats (32 KB)

    const int batch = in_sizes[0] / (IMG_W * IMG_H);  // 16 for the reference
    dim3 grid(IMG_W / TILE_W, IMG_H / TILE_H, batch); // (8, 32, 16) = 4096 blocks

    ssim_l1_partial_kernel<<<grid, NTHREADS, 0, stream>>>(pred, target, partials);

    const int nblocks = (int)(grid.x * grid.y * grid.z);
    const float inv_n = 1.0f / ((float)batch * (float)IMG_W * (float)IMG_H);
    ssim_l1_finalize_kernel<<<1, 256, 0, stream>>>(partials, nblocks, inv_n, out);
}